// BiDAF_53695681134720
// MI455X (gfx1250) — compile-verified
//
#include <hip/hip_runtime.h>
#include <hip/hip_bf16.h>
#include <math.h>

// Problem constants (from reference): B=32, L=768, D=768, L == D.
#define BB 32
#define LL 768
#define DD 768

typedef __attribute__((ext_vector_type(2))) float v2f;
typedef __attribute__((ext_vector_type(4))) float v4f;
typedef __attribute__((ext_vector_type(8))) float v8f;

// ---------------------------------------------------------------------------
// K1: v[b,d] = sum_j w3_w[j] * q[b,j,d]    (rank-1 contraction over q rows)
// grid = B * (D/256) = 96 blocks, 256 threads. Fully coalesced along d.
// ---------------------------------------------------------------------------
__global__ __launch_bounds__(256) void k1_compute_v(const float* __restrict__ q,
                                                    const float* __restrict__ w3_w,
                                                    float* __restrict__ v) {
  __shared__ float w[LL];
  const int b = blockIdx.x / (DD / 256);
  const int dchunk = blockIdx.x % (DD / 256);
  const int d = dchunk * 256 + threadIdx.x;
  for (int j = threadIdx.x; j < LL; j += 256) w[j] = w3_w[j];
  __syncthreads();
  const float* qb = q + (size_t)b * LL * DD;
  float acc = 0.f;
#pragma unroll 4
  for (int j = 0; j < LL; ++j) acc = fmaf(w[j], qb[(size_t)j * DD + d], acc);
  v[b * DD + d] = acc;
}

// ---------------------------------------------------------------------------
// K2: per-row dot products via V_WMMA_F32_16X16X4_F32.
//   h tiles: N=0 col -> w1h = h . w1_w ; N=1 col -> w3hq = h . v[b]
//   q tiles: N=0 col -> w2q = q . w2_w
// One wave per 16-row tile; 192 WMMAs of K=4 sweep the 768-long K dim.
//
// A layout (16x4 f32): lane m (<16): {A[m,k0],A[m,k0+1]}, lane 16+m: {A[m,k0+2],A[m,k0+3]}
// B layout (4x16 f32): VGPR0: K=0 (lanes0-15)/K=2 (lanes16-31); VGPR1: K=1/K=3, lane = N.
// C/D layout: VGPR r: lanes0-15 -> (M=r, N=lane); lanes16-31 -> (M=8+r, N=lane-16).
//
// All 32 lanes load B data unconditionally (D[:,n] depends only on B[:,n];
// accumulator columns we never read may hold garbage). This keeps the inner
// loop fully convergent: no saveexec/branches around the loads, just
// 2x global_load_b64 + v_wmma per iteration. EXEC stays all-1s as WMMA needs.
// ---------------------------------------------------------------------------
__global__ __launch_bounds__(128) void k2_dots_wmma(const float* __restrict__ h,
                                                    const float* __restrict__ q,
                                                    const float* __restrict__ w1_w,
                                                    const float* __restrict__ w2_w,
                                                    const float* __restrict__ v,
                                                    float* __restrict__ w1h,
                                                    float* __restrict__ w3hq,
                                                    float* __restrict__ w2q) {
  const int gtid = blockIdx.x * blockDim.x + threadIdx.x;
  const int wave = gtid >> 5;
  const int lane = threadIdx.x & 31;
  const int tiles_per_mat = BB * (LL / 16);  // 1536
  const bool is_q = wave >= tiles_per_mat;
  const int tile = is_q ? wave - tiles_per_mat : wave;
  const int b = tile / (LL / 16);
  const int i0 = (tile % (LL / 16)) * 16;

  const float* M = (is_q ? q : h) + (size_t)b * LL * DD + (size_t)i0 * DD;
  const float* W0 = is_q ? w2_w : w1_w;                 // column N=0 weights
  const float* W1 = is_q ? w2_w : (v + b * DD);          // column N=1 (q: dummy, unread)

  const int row   = lane & 15;       // M index this lane owns in A
  const int khalf = (lane >> 4) * 2; // 0 for lanes 0-15, 2 for lanes 16-31
  const int n     = lane & 15;       // N index this lane owns in B
  const float* rowp = M + (size_t)row * DD + khalf;
  // Lanes with n>=2 feed accumulator columns we never read; give them W0 so
  // every lane has a valid, convergent load (broadcast-friendly addresses).
  const float* wp = ((n == 1) ? W1 : W0) + khalf;

  v8f acc = {};
  for (int k0 = 0; k0 < DD; k0 += 4) {
    v2f a  = *(const v2f*)(rowp + k0);
    v2f bb = *(const v2f*)(wp + k0);
    acc = __builtin_amdgcn_wmma_f32_16x16x4_f32(false, a, false, bb,
                                                (short)0, acc, false, false);
  }

  const int mbase = (lane >> 4) * 8;  // lanes<16 carry M=0..7, lanes>=16 carry M=8..15
  float* out0 = is_q ? w2q : w1h;
  if ((lane & 15) == 0) {  // N=0 results live in lanes 0 and 16
    float* o = out0 + b * LL + i0 + mbase;
#pragma unroll
    for (int r = 0; r < 8; ++r) o[r] = acc[r];
  }
  if (!is_q && (lane & 15) == 1) {  // N=1 results live in lanes 1 and 17
    float* o = w3hq + b * LL + i0 + mbase;
#pragma unroll
    for (int r = 0; r < 8; ++r) o[r] = acc[r];
  }
}

// ---------------------------------------------------------------------------
// K3: per-batch softmaxes (768 wide). pj = softmax_j(w2q); p2 = softmax_i(w1h+w3hq).
// Biases cancel exactly under softmax shift-invariance, so they are never read.
// ---------------------------------------------------------------------------
__device__ __forceinline__ float block_max_768(float* arr, int t, float vv) {
  arr[t] = vv;
  __syncthreads();
  if (t < 256) arr[t] = fmaxf(arr[t], fmaxf(arr[t + 256], arr[t + 512]));
  __syncthreads();
  for (int s = 128; s > 0; s >>= 1) {
    if (t < s) arr[t] = fmaxf(arr[t], arr[t + s]);
    __syncthreads();
  }
  float r = arr[0];
  __syncthreads();
  return r;
}

__device__ __forceinline__ float block_sum_768(float* arr, int t, float vv) {
  arr[t] = vv;
  __syncthreads();
  if (t < 256) arr[t] = arr[t] + arr[t + 256] + arr[t + 512];
  __syncthreads();
  for (int s = 128; s > 0; s >>= 1) {
    if (t < s) arr[t] += arr[t + s];
    __syncthreads();
  }
  float r = arr[0];
  __syncthreads();
  return r;
}

__global__ __launch_bounds__(768) void k3_softmax(const float* __restrict__ w2q,
                                                  const float* __restrict__ w1h,
                                                  const float* __restrict__ w3hq,
                                                  float* __restrict__ pj,
                                                  float* __restrict__ p2) {
  __shared__ float arr[LL];
  const int b = blockIdx.x;
  const int t = threadIdx.x;

  // pj[b,j] = softmax_j(w2q[b,j])
  float s = w2q[b * LL + t];
  float mx = block_max_768(arr, t, s);
  float e = expf(s - mx);
  float sm = block_sum_768(arr, t, e);
  pj[b * LL + t] = e / sm;

  // p2[b,i] = softmax_i(w1h[b,i] + w3hq[b,i])
  float r = w1h[b * LL + t] + w3hq[b * LL + t];
  mx = block_max_768(arr, t, r);
  e = expf(r - mx);
  sm = block_sum_768(arr, t, e);
  p2[b * LL + t] = e / sm;
}

// ---------------------------------------------------------------------------
// K4: streaming output, bandwidth-bound: 302 MB of NT stores, 151 MB of reads
// (L2 hits — h+q were just touched and fit in the 192 MB L2).
//   out[b, 0L+i, :] = h
//   out[b, 1L+i, :] = c  = q * pj          (pj broadcast along rows)
//   out[b, 2L+i, :] = h * c
//   out[b, 3L+i, :] = (h*p2[b,i]) * c
// One block per (b,i) row; 192 threads x float4 covers D=768.
// ---------------------------------------------------------------------------
__global__ __launch_bounds__(192) void k4_output(const float* __restrict__ h,
                                                 const float* __restrict__ q,
                                                 const float* __restrict__ pj,
                                                 const float* __restrict__ p2,
                                                 float* __restrict__ out) {
  const int bi = blockIdx.x;        // b*L + i
  const int b = bi / LL;
  const int i = bi - b * LL;
  const int d = threadIdx.x * 4;

  const size_t rowoff = (size_t)bi * DD + d;
  const v4f h4  = *(const v4f*)(h + rowoff);
  const v4f q4  = *(const v4f*)(q + rowoff);
  const v4f pj4 = *(const v4f*)(pj + b * DD + d);
  const float p2s = p2[b * LL + i];

  const v4f c4 = q4 * pj4;
  const v4f hc4 = h4 * c4;
  const v4f qcc4 = hc4 * p2s;

  const size_t LD = (size_t)LL * DD;
  float* ob = out + ((size_t)b * 4 * LL + i) * DD + d;
  __builtin_nontemporal_store(h4,   (v4f*)(ob));
  __builtin_nontemporal_store(c4,   (v4f*)(ob + LD));
  __builtin_nontemporal_store(hc4,  (v4f*)(ob + 2 * LD));
  __builtin_nontemporal_store(qcc4, (v4f*)(ob + 3 * LD));
}

// ---------------------------------------------------------------------------
// Launcher. Input order: h, q, w1_w, w1_b, w2_w, w2_b, w3_w, w3_b (biases
// provably cancel under both softmaxes and never appear elsewhere).
// ---------------------------------------------------------------------------
extern "C" void kernel_launch(void* const* d_in, const int* in_sizes, int n_in,
                              void* d_out, int out_size, void* d_ws, size_t ws_size,
                              hipStream_t stream) {
  const float* h    = (const float*)d_in[0];
  const float* q    = (const float*)d_in[1];
  const float* w1_w = (const float*)d_in[2];
  const float* w2_w = (const float*)d_in[4];
  const float* w3_w = (const float*)d_in[6];
  float* out = (float*)d_out;

  // Workspace layout (floats): v | w1h | w3hq | w2q | pj | p2  (each B*768)
  float* ws   = (float*)d_ws;
  const int S = BB * LL;  // 24576
  float* v    = ws;
  float* w1h  = ws + 1 * S;
  float* w3hq = ws + 2 * S;
  float* w2q  = ws + 3 * S;
  float* pj   = ws + 4 * S;
  float* p2   = ws + 5 * S;

  // K1: v[b] = w3_w^T @ q[b]
  k1_compute_v<<<BB * (DD / 256), 256, 0, stream>>>(q, w3_w, v);

  // K2: WMMA GEMVs. waves = 2 * B * (L/16) = 3072 -> 768 blocks x 128 threads.
  k2_dots_wmma<<<(2 * BB * (LL / 16) * 32) / 128, 128, 0, stream>>>(
      h, q, w1_w, w2_w, v, w1h, w3hq, w2q);

  // K3: per-batch softmaxes
  k3_softmax<<<BB, LL, 0, stream>>>(w2q, w1h, w3hq, pj, p2);

  // K4: streaming output
  k4_output<<<BB * LL, DD / 4, 0, stream>>>(h, q, pj, p2, out);
}